// MatrixFactorizationIF_31095563223421
// MI455X (gfx1250) — compile-verified
//
#include <hip/hip_runtime.h>
#include <hip/hip_bf16.h>

typedef __attribute__((ext_vector_type(2))) float v2f;
typedef __attribute__((ext_vector_type(8))) float v8f;

#define R_DIM 64
#define S_DIM 3
#define ROW_F 448            // R*(1+2S)
#define ALPHA_C 0.001f
#define BETA_C 0.001f

// Select element s (0..7) out of an 8-VGPR accumulator without scratch:
// lowers to a short v_cndmask chain.
__device__ __forceinline__ float pick8(v8f d, int s) {
    float r = d[0];
    r = (s == 1) ? d[1] : r;
    r = (s == 2) ? d[2] : r;
    r = (s == 3) ? d[3] : r;
    r = (s == 4) ? d[4] : r;
    r = (s == 5) ? d[5] : r;
    r = (s == 6) ? d[6] : r;
    r = (s == 7) ? d[7] : r;
    return r;
}

__global__ void __launch_bounds__(256)
mf_if_wmma_kernel(const float* __restrict__ pF,
                  const float* __restrict__ M,
                  const int*   __restrict__ ijk,
                  float*       __restrict__ out,
                  int B)
{
    const int tid  = blockIdx.x * blockDim.x + threadIdx.x;
    const int wave = tid >> 5;               // wave32
    const int lane = threadIdx.x & 31;
    const int m    = lane & 15;              // element slot / matrix row+col
    const bool hi  = lane >= 16;             // holds K={2,3} half of the chunk

    const int b_raw = wave * 16 + m;
    const int b     = (b_raw < B) ? b_raw : (B - 1);   // clamp: EXEC must stay full for WMMA

    const int* t = ijk + 3 * (size_t)b;
    const int i  = t[0];
    const int j  = t[1];
    const int k  = t[2];
    const int kc = (k < 0) ? 0 : k;
    const float mask = (k != -1) ? 1.0f : 0.0f;

    const float* pFi = pF + (size_t)i * ROW_F;
    const float* Mj  = M + (size_t)j  * R_DIM;
    const float* Mk  = M + (size_t)kc * R_DIM;

    const int koff = hi ? 2 : 0;     // K offset within the 4-wide chunk
    const int voff = hi ? 6 : 0;     // offset into the interleaved (r,S) V blocks

    v8f dP  = {};
    v8f dA0 = {}, dA1 = {}, dA2 = {};
    v8f dG0 = {}, dG1 = {}, dG2 = {};

    #pragma unroll
    for (int c = 0; c < 16; ++c) {
        const int kb = 4 * c + koff;

        v2f aP = *(const v2f*)(pFi + kb);    // Pi chunk   (A row m, K pair)
        v2f bJ = *(const v2f*)(Mj  + kb);    // Mj chunk   (B col m, K pair)
        v2f bK = *(const v2f*)(Mk  + kb);    // Mk chunk

        // Vs / Vg are (r,S)-interleaved: element (r,s) at [base + 3*r + s].
        // This lane covers r = 4c+koff and 4c+koff+1 -> 6 consecutive floats.
        const float* vs = pFi + R_DIM + 12 * c + voff;
        const float* vg = pFi + (1 + S_DIM) * R_DIM + 12 * c + voff;
        v2f s01 = *(const v2f*)(vs + 0);
        v2f s23 = *(const v2f*)(vs + 2);
        v2f s45 = *(const v2f*)(vs + 4);
        v2f g01 = *(const v2f*)(vg + 0);
        v2f g23 = *(const v2f*)(vg + 2);
        v2f g45 = *(const v2f*)(vg + 4);

        v2f aS0 = { s01.x, s23.y };          // Vs[:,0] pair
        v2f aS1 = { s01.y, s45.x };          // Vs[:,1] pair
        v2f aS2 = { s23.x, s45.y };          // Vs[:,2] pair
        v2f aG0 = { g01.x, g23.y };
        v2f aG1 = { g01.y, g45.x };
        v2f aG2 = { g23.x, g45.y };

        dP  = __builtin_amdgcn_wmma_f32_16x16x4_f32(false, aP,  false, bJ, (short)0, dP,  false, false);
        dA0 = __builtin_amdgcn_wmma_f32_16x16x4_f32(false, aS0, false, bJ, (short)0, dA0, false, false);
        dA1 = __builtin_amdgcn_wmma_f32_16x16x4_f32(false, aS1, false, bJ, (short)0, dA1, false, false);
        dA2 = __builtin_amdgcn_wmma_f32_16x16x4_f32(false, aS2, false, bJ, (short)0, dA2, false, false);
        dG0 = __builtin_amdgcn_wmma_f32_16x16x4_f32(false, aG0, false, bK, (short)0, dG0, false, false);
        dG1 = __builtin_amdgcn_wmma_f32_16x16x4_f32(false, aG1, false, bK, (short)0, dG1, false, false);
        dG2 = __builtin_amdgcn_wmma_f32_16x16x4_f32(false, aG2, false, bK, (short)0, dG2, false, false);
    }

    // Diagonal extraction:
    //  n < 8  : D[n,n] in lane n,     VGPR n
    //  n >= 8 : D[n,n] in lane n+16,  VGPR n-8
    const bool holder = (lane < 8) || (lane >= 24);
    const int  sel    = (lane < 8) ? lane : (lane - 24);

    const float p  = pick8(dP,  sel);
    const float a0 = pick8(dA0, sel);
    const float a1 = pick8(dA1, sel);
    const float a2 = pick8(dA2, sel);
    const float g0 = pick8(dG0, sel);
    const float g1 = pick8(dG1, sel);
    const float g2 = pick8(dG2, sel);

    const float res = ALPHA_C * p +
                      mask * (BETA_C * BETA_C) * (a0 * g0 + a1 * g1 + a2 * g2);

    if (holder && b_raw < B) {
        out[b_raw] = res;   // holder lane's element index == lane & 15 == n
    }
}

extern "C" void kernel_launch(void* const* d_in, const int* in_sizes, int n_in,
                              void* d_out, int out_size, void* d_ws, size_t ws_size,
                              hipStream_t stream)
{
    (void)n_in; (void)d_ws; (void)ws_size; (void)out_size;
    const float* pF  = (const float*)d_in[0];
    const float* M   = (const float*)d_in[1];
    const int*   ijk = (const int*)d_in[2];
    float*       out = (float*)d_out;

    const int B = in_sizes[2] / 3;
    const int waves   = (B + 15) / 16;       // one wave32 per 16 batch elements
    const int threads = waves * 32;
    const int block   = 256;                 // 8 waves per block
    const int grid    = (threads + block - 1) / block;

    mf_if_wmma_kernel<<<grid, block, 0, stream>>>(pF, M, ijk, out, B);
}